// CustomLayer_55027120996497
// MI455X (gfx1250) — compile-verified
//
#include <hip/hip_runtime.h>
#include <hip/hip_bf16.h>

// GIN layer: scatter-add (memory bound, ~1GB traffic) + 2-layer MLP via
// fp32 WMMA (v_wmma_f32_16x16x4_f32), fused bias/ReLU/(1+eps)*h update.

#define EMB 96
#define HID 192

typedef __attribute__((ext_vector_type(2))) float v2f;
typedef __attribute__((ext_vector_type(8))) float v8f;

__global__ __launch_bounds__(256) void gin_zero_kernel(float* __restrict__ p, int n) {
    int i = blockIdx.x * blockDim.x + threadIdx.x;
    if (i < n) p[i] = 0.0f;
}

// One thread per (edge, 16-byte feature chunk): 96 floats = 24 float4 chunks.
__global__ __launch_bounds__(256) void gin_scatter_kernel(
    const float* __restrict__ h, const float* __restrict__ edge_attr,
    const int* __restrict__ src, const int* __restrict__ dst,
    float* __restrict__ aggr, int n_edges)
{
    long long tid = (long long)blockIdx.x * blockDim.x + threadIdx.x;
    int e = (int)(tid / 24);
    int c = (int)(tid % 24);
    if (e >= n_edges) return;

    int s = src[e];
    int d = dst[e];
    const float4* hs = (const float4*)(h + (size_t)s * EMB);
    const float4* es = (const float4*)(edge_attr + (size_t)e * EMB);
    float4 hv = hs[c];
    float4 ev = es[c];
    float* o = aggr + (size_t)d * EMB + c * 4;
    atomicAdd(o + 0, hv.x + ev.x);   // non-returning global_atomic_add_f32
    atomicAdd(o + 1, hv.y + ev.y);
    atomicAdd(o + 2, hv.z + ev.z);
    atomicAdd(o + 3, hv.w + ev.w);
}

// One block = 16 nodes, 6 waves (192 threads, wave32).
// `out` holds aggr on entry (this block's rows are staged to LDS before being
// overwritten), final GIN output on exit.
__global__ __launch_bounds__(192) void gin_mlp_kernel(
    const float* __restrict__ h,
    const float* __restrict__ W1, const float* __restrict__ b1,
    const float* __restrict__ W2, const float* __restrict__ b2,
    const float* __restrict__ eps,
    float* __restrict__ out, int n_nodes)
{
    __shared__ float aggrS[16][EMB];
    __shared__ float hidS[16][HID];

    const int row0 = blockIdx.x * 16;
    const int tid  = threadIdx.x;      // 0..191
    const int wave = tid >> 5;         // 0..5
    const int lane = tid & 31;
    const int lrow = lane & 15;        // A row / C-D column
    const int lhi  = lane >> 4;        // 0 or 1 (K-half / M-half select)

    // Stage this block's aggr tile (16 x 96) into LDS.
    for (int i = tid; i < 16 * EMB; i += 192) {
        int r = i / EMB, c = i % EMB;
        int node = row0 + r;
        aggrS[r][c] = (node < n_nodes) ? out[(size_t)node * EMB + c] : 0.0f;
    }
    __syncthreads();

    // ---- Layer 1: hidden[16x192] = relu(aggr @ W1 + b1). 12 tiles, 2/wave.
    for (int t = 0; t < 2; ++t) {
        int nt   = wave + t * 6;             // 0..11
        int ncol = nt * 16 + lrow;           // column in HID
        v8f acc = {};
        for (int k = 0; k < EMB; k += 4) {
            int kb = k + 2 * lhi;
            v2f a, b;
            a.x = aggrS[lrow][kb];
            a.y = aggrS[lrow][kb + 1];
            b.x = W1[(size_t)kb * HID + ncol];
            b.y = W1[(size_t)(kb + 1) * HID + ncol];
            acc = __builtin_amdgcn_wmma_f32_16x16x4_f32(
                false, a, false, b, (short)0, acc, false, false);
        }
        float bias = b1[ncol];
        #pragma unroll
        for (int i = 0; i < 8; ++i) {
            float v = acc[i] + bias;
            hidS[i + 8 * lhi][ncol] = v > 0.0f ? v : 0.0f;
        }
    }
    __syncthreads();

    // ---- Layer 2 + GIN update: out = (1+eps)*h + hidden @ W2 + b2. 6 tiles, 1/wave.
    {
        int ncol = wave * 16 + lrow;         // column in EMB
        v8f acc = {};
        for (int k = 0; k < HID; k += 4) {
            int kb = k + 2 * lhi;
            v2f a, b;
            a.x = hidS[lrow][kb];
            a.y = hidS[lrow][kb + 1];
            b.x = W2[(size_t)kb * EMB + ncol];
            b.y = W2[(size_t)(kb + 1) * EMB + ncol];
            acc = __builtin_amdgcn_wmma_f32_16x16x4_f32(
                false, a, false, b, (short)0, acc, false, false);
        }
        float bias  = b2[ncol];
        float scale = 1.0f + eps[0];
        #pragma unroll
        for (int i = 0; i < 8; ++i) {
            int node = row0 + i + 8 * lhi;
            if (node < n_nodes) {
                size_t idx = (size_t)node * EMB + ncol;
                out[idx] = scale * h[idx] + acc[i] + bias;
            }
        }
    }
}

extern "C" void kernel_launch(void* const* d_in, const int* in_sizes, int n_in,
                              void* d_out, int out_size, void* d_ws, size_t ws_size,
                              hipStream_t stream) {
    const float* h    = (const float*)d_in[0];
    const float* ea   = (const float*)d_in[1];
    const int*   src  = (const int*)d_in[2];
    const int*   dst  = (const int*)d_in[3];
    const float* W1   = (const float*)d_in[4];
    const float* b1   = (const float*)d_in[5];
    const float* W2   = (const float*)d_in[6];
    const float* b2   = (const float*)d_in[7];
    const float* eps  = (const float*)d_in[8];
    float* out = (float*)d_out;

    const int n_nodes = in_sizes[0] / EMB;   // 50000
    const int n_edges = in_sizes[2];         // 800000
    const int n_aggr  = n_nodes * EMB;

    // d_out doubles as the aggregation buffer (same shape as output).
    gin_zero_kernel<<<(n_aggr + 255) / 256, 256, 0, stream>>>(out, n_aggr);

    long long chunks = (long long)n_edges * 24;
    int sblocks = (int)((chunks + 255) / 256);
    gin_scatter_kernel<<<sblocks, 256, 0, stream>>>(h, ea, src, dst, out, n_edges);

    int mblocks = (n_nodes + 15) / 16;       // 3125
    gin_mlp_kernel<<<mblocks, 192, 0, stream>>>(h, W1, b1, W2, b2, eps, out, n_nodes);
}